// SMART_1m_22316650070987
// MI455X (gfx1250) — compile-verified
//
#include <hip/hip_runtime.h>
#include <hip/hip_bf16.h>

// ---------------------------------------------------------------------------
// GraphSAGE autoencoder for MI455X (gfx1250, wave32).
//   - edge scatter-aggregation: hardware fp32 global atomics (bandwidth bound)
//   - dense layers: V_WMMA_F32_16X16X4_F32 (fp32 matrix pipe, matches ref fp32)
//   - row L2 normalize: wave shuffle reduction
// ---------------------------------------------------------------------------

#define N_NODES 50000
#define N_EDGES 800000
#define IN_DIM  256
#define OUT_DIM 128

typedef __attribute__((ext_vector_type(2))) float v2f;
typedef __attribute__((ext_vector_type(8))) float v8f;

// ---------------------------------------------------------------------------
// Degree / inverse degree
// ---------------------------------------------------------------------------
__global__ void deg_kernel(const int* __restrict__ dst, float* __restrict__ deg,
                           int n_edges) {
    int i = blockIdx.x * blockDim.x + threadIdx.x;
    if (i < n_edges) {
        unsafeAtomicAdd(&deg[dst[i]], 1.0f);   // global_atomic_add_f32
    }
}

__global__ void invdeg_kernel(float* __restrict__ deg, int n) {
    int i = blockIdx.x * blockDim.x + threadIdx.x;
    if (i < n) deg[i] = 1.0f / fmaxf(deg[i], 1.0f);
}

// ---------------------------------------------------------------------------
// Scatter: agg[dst] += x[src]   (one wave per edge, lanes stride the feature)
// ---------------------------------------------------------------------------
template <int DIN>
__global__ void scatter_add_kernel(const float* __restrict__ x,
                                   const int* __restrict__ src,
                                   const int* __restrict__ dst,
                                   float* __restrict__ agg, int n_edges) {
    int wave = (blockIdx.x * blockDim.x + threadIdx.x) >> 5;
    int lane = threadIdx.x & 31;
    if (wave >= n_edges) return;
    int s = src[wave];
    int d = dst[wave];
    const float* __restrict__ xr = x + (size_t)s * DIN;
    float* __restrict__ ar = agg + (size_t)d * DIN;
#pragma unroll
    for (int j = lane; j < DIN; j += 32) {
        unsafeAtomicAdd(&ar[j], xr[j]);        // global_atomic_add_f32
    }
}

// ---------------------------------------------------------------------------
// Fused SAGE GEMM:  out = (agg * inv_deg) @ Wl^T + x @ Wr^T + bias
// (HAS_AGG=false -> plain  out = x @ Wr^T + bias, used for the FC layer)
//
// One wave computes one 16x16 output tile with V_WMMA_F32_16X16X4_F32.
// A 16x4 fp32 operand layout (2 VGPRs): lanes 0-15 rows M, K={k,k+1};
// lanes 16-31 same rows, K={k+2,k+3}. B 4x16 mirrors with N across lanes.
// ---------------------------------------------------------------------------
template <int DIN, bool HAS_AGG>
__global__ void sage_gemm_wmma_kernel(const float* __restrict__ agg,
                                      const float* __restrict__ x,
                                      const float* __restrict__ inv_deg,
                                      const float* __restrict__ Wl,  // [dout, DIN]
                                      const float* __restrict__ Wr,  // [dout, DIN]
                                      const float* __restrict__ bias,
                                      float* __restrict__ out,
                                      int n_rows, int dout) {
    const int wave = threadIdx.x >> 5;               // 4 waves / block
    const int lane = threadIdx.x & 31;
    const int m_tile = blockIdx.x * 4 + wave;        // wave-uniform
    if (m_tile * 16 >= n_rows) return;               // uniform exit: EXEC stays full
    const int n_base = blockIdx.y * 16;

    const int lp   = lane & 15;                      // row/col within tile
    const int koff = (lane >> 4) * 2;                // K sub-offset per half-wave

    const int row = m_tile * 16 + lp;                // A row for this lane
    const int col = n_base + lp;                     // B column for this lane

    const float* __restrict__ xRow  = x + (size_t)row * DIN;
    const float* __restrict__ wrRow = Wr + (size_t)col * DIN;
    const float* __restrict__ aggRow = HAS_AGG ? agg + (size_t)row * DIN : nullptr;
    const float* __restrict__ wlRow  = HAS_AGG ? Wl + (size_t)col * DIN : nullptr;
    const float scale = HAS_AGG ? inv_deg[row] : 0.0f;

    v8f c = {};
#pragma unroll
    for (int k = 0; k < DIN; k += 4) {
        if (HAS_AGG) {
            v2f a1 = *(const v2f*)(aggRow + k + koff);
            a1 = a1 * scale;                          // fold inv_deg into A
            v2f b1 = *(const v2f*)(wlRow + k + koff);
            c = __builtin_amdgcn_wmma_f32_16x16x4_f32(
                    false, a1, false, b1, (short)0, c, false, false);
        }
        v2f a2 = *(const v2f*)(xRow + k + koff);
        v2f b2 = *(const v2f*)(wrRow + k + koff);
        c = __builtin_amdgcn_wmma_f32_16x16x4_f32(
                false, a2, false, b2, (short)0, c, false, false);
    }

    const float bv = bias[col];
    const int half = lane >> 4;
#pragma unroll
    for (int v = 0; v < 8; ++v) {
        int m = m_tile * 16 + v + 8 * half;           // C/D layout: VGPR v, halves
        out[(size_t)m * dout + col] = c[v] + bv;
    }
}

// ---------------------------------------------------------------------------
// Row-wise L2 normalize (one wave per row)
// ---------------------------------------------------------------------------
__global__ void l2norm_kernel(const float* __restrict__ in, float* __restrict__ out,
                              int n_rows, int d) {
    int row  = blockIdx.x * (blockDim.x >> 5) + (threadIdx.x >> 5);
    int lane = threadIdx.x & 31;
    if (row >= n_rows) return;
    const float* __restrict__ r = in + (size_t)row * d;
    float s = 0.0f;
    for (int j = lane; j < d; j += 32) { float v = r[j]; s += v * v; }
#pragma unroll
    for (int off = 16; off > 0; off >>= 1) s += __shfl_xor(s, off, 32);
    float inv = 1.0f / fmaxf(sqrtf(s), 1e-12f);
    float* __restrict__ o = out + (size_t)row * d;
    for (int j = lane; j < d; j += 32) o[j] = r[j] * inv;
}

// ---------------------------------------------------------------------------
// Host orchestration
// ---------------------------------------------------------------------------
extern "C" void kernel_launch(void* const* d_in, const int* in_sizes, int n_in,
                              void* d_out, int out_size, void* d_ws, size_t ws_size,
                              hipStream_t stream) {
    const float* features = (const float*)d_in[0];
    const int*   edge     = (const int*)d_in[1];
    const int*   e_src    = edge;
    const int*   e_dst    = edge + N_EDGES;
    const float* Wl_e1 = (const float*)d_in[2];
    const float* bl_e1 = (const float*)d_in[3];
    const float* Wr_e1 = (const float*)d_in[4];
    const float* Wl_e2 = (const float*)d_in[5];
    const float* bl_e2 = (const float*)d_in[6];
    const float* Wr_e2 = (const float*)d_in[7];
    const float* W_fc  = (const float*)d_in[8];
    const float* b_fc  = (const float*)d_in[9];
    const float* Wl_d1 = (const float*)d_in[10];
    const float* bl_d1 = (const float*)d_in[11];
    const float* Wr_d1 = (const float*)d_in[12];
    const float* Wl_d2 = (const float*)d_in[13];
    const float* bl_d2 = (const float*)d_in[14];
    const float* Wr_d2 = (const float*)d_in[15];

    float* x1_out  = (float*)d_out;                       // [N, 128]
    float* rec_out = (float*)d_out + (size_t)N_NODES * OUT_DIM; // [N, 256]

    // workspace carve (floats)
    float* ws   = (float*)d_ws;
    float* deg  = ws;                         ws += N_NODES;            // -> inv_deg
    float* agg  = ws;                         ws += (size_t)N_NODES * IN_DIM;
    float* tmp  = ws;                         ws += (size_t)N_NODES * IN_DIM;
    float* bufA = ws;                         ws += (size_t)N_NODES * OUT_DIM;
    float* bufB = ws;                         ws += (size_t)N_NODES * OUT_DIM;

    const int scat_blocks = (N_EDGES * 32 + 255) / 256;   // 1 wave per edge
    const dim3 gemm_blk(128);
    const int  m_blocks = (N_NODES / 16 + 3) / 4;         // 4 waves per block
    const dim3 gemm128(m_blocks, OUT_DIM / 16);
    const dim3 gemm256(m_blocks, IN_DIM / 16);
    const int norm_blocks = (N_NODES + 3) / 4;            // 4 rows per block

    // ---- degrees ----
    hipMemsetAsync(deg, 0, N_NODES * sizeof(float), stream);
    deg_kernel<<<(N_EDGES + 255) / 256, 256, 0, stream>>>(e_dst, deg, N_EDGES);
    invdeg_kernel<<<(N_NODES + 255) / 256, 256, 0, stream>>>(deg, N_NODES);

    // ---- encoder conv1: 256 -> 128 ----
    hipMemsetAsync(agg, 0, (size_t)N_NODES * IN_DIM * sizeof(float), stream);
    scatter_add_kernel<IN_DIM><<<scat_blocks, 256, 0, stream>>>(features, e_src, e_dst, agg, N_EDGES);
    sage_gemm_wmma_kernel<IN_DIM, true><<<gemm128, gemm_blk, 0, stream>>>(
        agg, features, deg, Wl_e1, Wr_e1, bl_e1, tmp, N_NODES, OUT_DIM);
    l2norm_kernel<<<norm_blocks, 128, 0, stream>>>(tmp, bufA, N_NODES, OUT_DIM);

    // ---- encoder conv2: 128 -> 128 ----
    hipMemsetAsync(agg, 0, (size_t)N_NODES * OUT_DIM * sizeof(float), stream);
    scatter_add_kernel<OUT_DIM><<<scat_blocks, 256, 0, stream>>>(bufA, e_src, e_dst, agg, N_EDGES);
    sage_gemm_wmma_kernel<OUT_DIM, true><<<gemm128, gemm_blk, 0, stream>>>(
        agg, bufA, deg, Wl_e2, Wr_e2, bl_e2, tmp, N_NODES, OUT_DIM);
    l2norm_kernel<<<norm_blocks, 128, 0, stream>>>(tmp, x1_out, N_NODES, OUT_DIM);

    // ---- bottleneck FC: x = x1 @ W_fc^T + b_fc ----
    sage_gemm_wmma_kernel<OUT_DIM, false><<<gemm128, gemm_blk, 0, stream>>>(
        nullptr, x1_out, nullptr, nullptr, W_fc, b_fc, bufB, N_NODES, OUT_DIM);

    // ---- decoder conv1: 128 -> 128 ----
    hipMemsetAsync(agg, 0, (size_t)N_NODES * OUT_DIM * sizeof(float), stream);
    scatter_add_kernel<OUT_DIM><<<scat_blocks, 256, 0, stream>>>(bufB, e_src, e_dst, agg, N_EDGES);
    sage_gemm_wmma_kernel<OUT_DIM, true><<<gemm128, gemm_blk, 0, stream>>>(
        agg, bufB, deg, Wl_d1, Wr_d1, bl_d1, tmp, N_NODES, OUT_DIM);
    l2norm_kernel<<<norm_blocks, 128, 0, stream>>>(tmp, bufA, N_NODES, OUT_DIM);

    // ---- decoder conv2: 128 -> 256 ----
    hipMemsetAsync(agg, 0, (size_t)N_NODES * OUT_DIM * sizeof(float), stream);
    scatter_add_kernel<OUT_DIM><<<scat_blocks, 256, 0, stream>>>(bufA, e_src, e_dst, agg, N_EDGES);
    sage_gemm_wmma_kernel<OUT_DIM, true><<<gemm256, gemm_blk, 0, stream>>>(
        agg, bufA, deg, Wl_d2, Wr_d2, bl_d2, tmp, N_NODES, IN_DIM);
    l2norm_kernel<<<norm_blocks, 128, 0, stream>>>(tmp, rec_out, N_NODES, IN_DIM);
}